// GCN_80891414052990
// MI455X (gfx1250) — compile-verified
//
#include <hip/hip_runtime.h>

#define DD 128
#define EPSBN 1e-5f

typedef __attribute__((ext_vector_type(2))) float v2f;
typedef __attribute__((ext_vector_type(4))) float v4f;
typedef __attribute__((ext_vector_type(8))) float v8f;

__device__ __forceinline__ void atomAddF(float* p, float v) {
  unsafeAtomicAdd(p, v);   // lowers to global_atomic_add_f32 (no CAS loop)
}

// ---------------------------------------------------------------------------
// WMMA fp32 GEMM: Out[M,128] = A[M,128] @ W[128,128]
// 128 threads (4 waves) per block; wave w owns rows blockIdx*64 + w*16 .. +15.
// W staged in LDS as pre-paired B fragments:
//   Wl2[(kk*2+hi)*288 + 2*col + j] = W[kk*4 + 2*hi + j][col]
// so each B fragment is a single 8B-aligned ds_load_b64 into an even VGPR
// pair (WMMA SRC alignment), and the 288-word stride (288%64==32) makes the
// full-wave b64 read touch all 64 LDS banks exactly once (conflict-free).
// ---------------------------------------------------------------------------
__global__ __launch_bounds__(128) void k_gemm128(const float* __restrict__ A,
                                                 const float* __restrict__ W,
                                                 float* __restrict__ Out,
                                                 int M) {
  __shared__ float Wl2[64 * 288];  // 73,728 B (WGP has 320 KB)
  const int tid = threadIdx.x;     // 128 threads: one column each for staging
  #pragma unroll 4
  for (int r = 0; r < DD; ++r) {
    int kk = r >> 2;
    int hi = (r >> 1) & 1;
    int j  = r & 1;
    Wl2[(kk * 2 + hi) * 288 + 2 * tid + j] = W[r * DD + tid];
  }
  __syncthreads();

  const int lane = tid & 31;
  const int wave = tid >> 5;
  const int lo   = lane & 15;
  const int hi   = lane >> 4;            // 0: K={0,1}, 1: K={2,3}
  const int row0 = blockIdx.x * 64 + wave * 16;
  if (row0 >= M) return;

  // A fragments for the whole K=128 strip: lane holds row (row0+lo),
  // columns kk*4 + 2*hi + {0,1}  -> one 8-byte load each.
  v2f a[32];
  const float* ap = A + (size_t)(row0 + lo) * DD + 2 * hi;
  #pragma unroll
  for (int kk = 0; kk < 32; ++kk) a[kk] = *(const v2f*)(ap + kk * 4);

  #pragma unroll
  for (int t = 0; t < 8; ++t) {
    v8f c = {0.f, 0.f, 0.f, 0.f, 0.f, 0.f, 0.f, 0.f};
    // B frag: lane holds col (t*16+lo), rows kk*4 + 2*hi + {0,1}
    const float* bp = &Wl2[hi * 288 + 2 * (t * 16 + lo)];
    #pragma unroll
    for (int kk = 0; kk < 32; ++kk) {
      v2f b = *(const v2f*)(bp + kk * 576);   // ds_load_b64, conflict-free
      c = __builtin_amdgcn_wmma_f32_16x16x4_f32(false, a[kk], false, b,
                                                (short)0, c, false, false);
    }
    // C layout: VGPR v -> row = row0 + v + 8*hi, col = t*16 + lo
    float* op = Out + (size_t)(row0 + 8 * hi) * DD + t * 16 + lo;
    #pragma unroll
    for (int v = 0; v < 8; ++v) op[(size_t)v * DD] = c[v];
  }
}

// ---------------------------------------------------------------------------
__global__ void k_fill(float* __restrict__ p, float v, int n) {
  int i = blockIdx.x * 256 + threadIdx.x;
  if (i < n) p[i] = v;
}

__global__ void k_deg(float* __restrict__ deg, const int* __restrict__ dst, int E) {
  int e = blockIdx.x * 256 + threadIdx.x;
  if (e < E) atomAddF(&deg[dst[e]], 1.0f);
}

__global__ void k_rsq(float* __restrict__ d, int n) {
  int i = blockIdx.x * 256 + threadIdx.x;
  if (i < n) d[i] = rsqrtf(d[i]);
}

// agg[i,d] = hW[i,d]*dinv[i]^2 + bias[d]   (self-loop term + conv bias)
__global__ void k_aggr_init(const float* __restrict__ hW, const float* __restrict__ dinv,
                            const float* __restrict__ bias, float* __restrict__ agg, int N) {
  size_t i = (size_t)blockIdx.x * 256 + threadIdx.x;
  if (i >= (size_t)N * DD) return;
  int row = (int)(i >> 7), d = (int)(i & 127);
  float di = dinv[row];
  agg[i] = hW[i] * di * di + bias[d];
}

// one wave per edge; lane handles 4 contiguous feature dims (float4 gather)
__global__ __launch_bounds__(256) void k_scatter(const float* __restrict__ hW,
                                                 const float* __restrict__ dinv,
                                                 const int* __restrict__ src,
                                                 const int* __restrict__ dst,
                                                 float* __restrict__ agg, int E) {
  size_t g = (size_t)blockIdx.x * 256 + threadIdx.x;
  int e = (int)(g >> 5);
  if (e >= E) return;
  int lane = (int)(g & 31);
  int s = src[e], d = dst[e];
  float norm = dinv[s] * dinv[d];
  v4f v = *(const v4f*)(hW + (size_t)s * DD + lane * 4);
  float* o = agg + (size_t)d * DD + lane * 4;
  atomAddF(o + 0, v.x * norm);
  atomAddF(o + 1, v.y * norm);
  atomAddF(o + 2, v.z * norm);
  atomAddF(o + 3, v.w * norm);
}

// ReLU (+ optional bias) in place, accumulate per-column sum/sumsq.
// Block handles 64 rows x 128 cols; thread (half, dcol) strides rows by 2.
__global__ __launch_bounds__(256) void k_relu_stats(float* __restrict__ h,
                                                    const float* __restrict__ bias,
                                                    float* __restrict__ sum,
                                                    float* __restrict__ sumsq, int M) {
  __shared__ float ls[256], lq[256];
  int tid = threadIdx.x;
  int dcol = tid & 127;
  int half = tid >> 7;
  int rowBase = blockIdx.x * 64;
  float b = bias ? bias[dcol] : 0.f;
  float s = 0.f, q = 0.f;
  for (int r = half; r < 64; r += 2) {
    int row = rowBase + r;
    if (row < M) {
      size_t idx = (size_t)row * DD + dcol;
      float v = h[idx] + b;
      v = v > 0.f ? v : 0.f;
      h[idx] = v;
      s += v;
      q += v * v;
    }
  }
  ls[tid] = s; lq[tid] = q;
  __syncthreads();
  if (tid < 128) {
    atomAddF(&sum[dcol],   ls[tid] + ls[tid + 128]);
    atomAddF(&sumsq[dcol], lq[tid] + lq[tid + 128]);
  }
}

__global__ void k_bn_finalize(const float* __restrict__ sum, const float* __restrict__ sumsq,
                              const float* __restrict__ g, const float* __restrict__ bb,
                              float* __restrict__ scale, float* __restrict__ shift,
                              float invM) {
  int d = threadIdx.x;
  float mu  = sum[d] * invM;
  float var = sumsq[d] * invM - mu * mu;
  float sc  = g[d] * rsqrtf(var + EPSBN);
  scale[d] = sc;
  shift[d] = bb[d] - mu * sc;
}

__global__ void k_bn_apply(float* __restrict__ h, const float* __restrict__ scale,
                           const float* __restrict__ shift, size_t n) {
  size_t i = (size_t)blockIdx.x * 256 + threadIdx.x;
  if (i >= n) return;
  int d = (int)(i & 127);
  h[i] = h[i] * scale[d] + shift[d];
}

__global__ void k_pool(const float* __restrict__ h, const int* __restrict__ batch,
                       float* __restrict__ pool, int N) {
  size_t i = (size_t)blockIdx.x * 256 + threadIdx.x;
  if (i >= (size_t)N * DD) return;
  int row = (int)(i >> 7);
  atomAddF(&pool[(size_t)batch[row] * DD + (i & 127)], h[i]);
}

// one wave per graph: dot(h[g,:], out_W) + out_b
__global__ __launch_bounds__(256) void k_head(const float* __restrict__ h,
                                              const float* __restrict__ ow,
                                              const float* __restrict__ ob,
                                              float* __restrict__ out, int G) {
  size_t gt = (size_t)blockIdx.x * 256 + threadIdx.x;
  int g = (int)(gt >> 5);
  if (g >= G) return;
  int lane = (int)(gt & 31);
  const float* hp = h + (size_t)g * DD + lane * 4;
  const float* wp = ow + lane * 4;
  float p = hp[0] * wp[0] + hp[1] * wp[1] + hp[2] * wp[2] + hp[3] * wp[3];
  #pragma unroll
  for (int off = 16; off > 0; off >>= 1) p += __shfl_down(p, off, 32);
  if (lane == 0) out[g] = p + ob[0];
}

// ---------------------------------------------------------------------------
extern "C" void kernel_launch(void* const* d_in, const int* in_sizes, int n_in,
                              void* d_out, int out_size, void* d_ws, size_t ws_size,
                              hipStream_t stream) {
  const int N = in_sizes[0] / DD;
  const int E = in_sizes[1] / 2;
  const int G = out_size;  // [G,1]

  const float* x      = (const float*)d_in[0];
  const int*   esrc   = (const int*)d_in[1];
  const int*   edst   = esrc + E;
  const int*   batch  = (const int*)d_in[2];
  const float* convW  = (const float*)d_in[3];
  const float* convB  = (const float*)d_in[4];
  const float* convG  = (const float*)d_in[5];
  const float* convBb = (const float*)d_in[6];
  const float* hidW   = (const float*)d_in[7];
  const float* hidB   = (const float*)d_in[8];
  const float* hidG   = (const float*)d_in[9];
  const float* hidBb  = (const float*)d_in[10];
  const float* outW   = (const float*)d_in[11];
  const float* outB   = (const float*)d_in[12];

  char* p = (char*)d_ws;
  float* bufA  = (float*)p; p += (size_t)N * DD * 4;   // h / agg
  float* bufB  = (float*)p; p += (size_t)N * DD * 4;   // h @ W
  float* dinv  = (float*)p; p += (size_t)N * 4;
  float* poolA = (float*)p; p += (size_t)G * DD * 4;
  float* poolB = (float*)p; p += (size_t)G * DD * 4;
  float* stats = (float*)p;                             // [sum|sumsq|scale|shift] 4*128

  const int ND_blk = (int)(((size_t)N * DD + 255) / 256);
  const int GD_blk = (G * DD + 255) / 256;

  // degree -> 1/sqrt(deg)
  k_fill<<<(N + 255) / 256, 256, 0, stream>>>(dinv, 1.0f, N);
  k_deg <<<(E + 255) / 256, 256, 0, stream>>>(dinv, edst, E);
  k_rsq <<<(N + 255) / 256, 256, 0, stream>>>(dinv, N);

  const float* hcur = x;
  for (int l = 0; l < 3; ++l) {
    k_gemm128<<<(N + 63) / 64, 128, 0, stream>>>(hcur, convW + l * DD * DD, bufB, N);
    k_aggr_init<<<ND_blk, 256, 0, stream>>>(bufB, dinv, convB + l * DD, bufA, N);
    k_scatter<<<(int)(((size_t)E * 32 + 255) / 256), 256, 0, stream>>>(bufB, dinv, esrc, edst, bufA, E);
    k_fill<<<1, 256, 0, stream>>>(stats, 0.f, 2 * DD);
    k_relu_stats<<<(N + 63) / 64, 256, 0, stream>>>(bufA, nullptr, stats, stats + DD, N);
    k_bn_finalize<<<1, 128, 0, stream>>>(stats, stats + DD, convG + l * DD, convBb + l * DD,
                                         stats + 2 * DD, stats + 3 * DD, 1.0f / (float)N);
    k_bn_apply<<<ND_blk, 256, 0, stream>>>(bufA, stats + 2 * DD, stats + 3 * DD, (size_t)N * DD);
    hcur = bufA;
  }

  // global_add_pool
  k_fill<<<GD_blk, 256, 0, stream>>>(poolA, 0.f, G * DD);
  k_pool<<<ND_blk, 256, 0, stream>>>(bufA, batch, poolA, N);

  float* pin = poolA;
  float* pout = poolB;
  for (int l = 0; l < 2; ++l) {
    k_gemm128<<<(G + 63) / 64, 128, 0, stream>>>(pin, hidW + l * DD * DD, pout, G);
    k_fill<<<1, 256, 0, stream>>>(stats, 0.f, 2 * DD);
    k_relu_stats<<<(G + 63) / 64, 256, 0, stream>>>(pout, hidB + l * DD, stats, stats + DD, G);
    k_bn_finalize<<<1, 128, 0, stream>>>(stats, stats + DD, hidG + l * DD, hidBb + l * DD,
                                         stats + 2 * DD, stats + 3 * DD, 1.0f / (float)G);
    k_bn_apply<<<GD_blk, 256, 0, stream>>>(pout, stats + 2 * DD, stats + 3 * DD, (size_t)G * DD);
    float* t = pin; pin = pout; pout = t;
  }

  k_head<<<(int)(((size_t)G * 32 + 255) / 256), 256, 0, stream>>>(pin, outW, outB, (float*)d_out, G);
}